// RoiAlign_79130477461990
// MI455X (gfx1250) — compile-verified
//
#include <hip/hip_runtime.h>

// RoIAlign on MI455X (gfx1250, wave32).
//
// Roofline: output = 50.2 MB fp32 stores, feature map 6.55 MB (L2-resident),
// compute ~0.5 GFLOP -> store-bandwidth bound (~2-3 us floor @ 23.3 TB/s).
// WMMA recast would inflate work ~40000x -> rejected; this is a gather op.
// Lane = 4 consecutive channels (ext-vector float4) -> all gathers/stores are
// B128 (4x fewer vmem instructions, 4x less address VALU per byte). Per-box
// sampling blob staged via CDNA5 async-tensor path (ASYNCcnt). NT stores
// keep the output stream from evicting the feature map in L2.

typedef float v4f __attribute__((ext_vector_type(4)));

#define RA_P    7
#define RA_SR   2
#define RA_C    256
#define RA_H    80
#define RA_W    80
#define RA_NS   (RA_P * RA_SR)          // 14 samples per axis
#define RA_HW   (RA_H * RA_W)           // 6400
#define RA_BINS (RA_P * RA_P)           // 49
#define RA_SCALE (1.0f / 32.0f)

// ---------------------------------------------------------------------------
// Kernel 1: CHW -> HWC transpose (LDS tiled, conflict-free 32x33 tile).
// ---------------------------------------------------------------------------
__global__ __launch_bounds__(256) void ra_chw_to_hwc(const float* __restrict__ in,
                                                     float* __restrict__ out) {
    __shared__ float tile[32][33];
    const int hw0 = blockIdx.x * 32;   // 200 blocks
    const int c0  = blockIdx.y * 32;   // 8 blocks
    const int tx = threadIdx.x;        // 0..31
    const int ty = threadIdx.y;        // 0..7
#pragma unroll
    for (int k = 0; k < 4; ++k) {
        const int c = c0 + ty + 8 * k;
        tile[ty + 8 * k][tx] = in[(size_t)c * RA_HW + hw0 + tx];
    }
    __syncthreads();
#pragma unroll
    for (int k = 0; k < 4; ++k) {
        const int hw = hw0 + ty + 8 * k;
        out[(size_t)hw * RA_C + c0 + tx] = tile[tx][ty + 8 * k];
    }
}

// ---------------------------------------------------------------------------
// Kernel 2: per-box sampling tables (matches reference _axis_coords exactly).
// Blob per box: 128 dwords (512 B, one wave32 x B128 async load):
//   [0..13] ylo  [14..27] yhi  [28..41] wyl  [42..55] wyh
//   [64..77] xlo [78..91] xhi  [92..105] wxl [106..119] wxh
// ---------------------------------------------------------------------------
__device__ __forceinline__ void ra_axis(float start, float binsz, float size,
                                        unsigned* lo, unsigned* hi,
                                        unsigned* wl, unsigned* wh) {
#pragma unroll
    for (int j = 0; j < RA_NS; ++j) {
        const float off = ((float)j + 0.5f) / (float)RA_SR;
        const float pos = start + binsz * off;
        const float v   = (pos >= -1.0f && pos <= size) ? 1.0f : 0.0f;
        const float p   = fminf(fmaxf(pos, 0.0f), size - 1.0f);
        const float lof = floorf(p);
        const float hif = fminf(lof + 1.0f, size - 1.0f);
        const float whv = p - lof;
        const float wlv = 1.0f - whv;
        lo[j] = (unsigned)(int)lof;
        hi[j] = (unsigned)(int)hif;
        wl[j] = __float_as_uint(wlv * v);
        wh[j] = __float_as_uint(whv * v);
    }
}

__global__ __launch_bounds__(128) void ra_coords(const float* __restrict__ props,
                                                 unsigned* __restrict__ blobs,
                                                 int n_boxes) {
    const int n = blockIdx.x * blockDim.x + threadIdx.x;
    if (n >= n_boxes) return;
    const float x1 = props[n * 4 + 0] * RA_SCALE;
    const float y1 = props[n * 4 + 1] * RA_SCALE;
    const float x2 = props[n * 4 + 2] * RA_SCALE;
    const float y2 = props[n * 4 + 3] * RA_SCALE;
    const float bw = fmaxf(x2 - x1, 1.0f) / (float)RA_P;
    const float bh = fmaxf(y2 - y1, 1.0f) / (float)RA_P;
    unsigned* b = blobs + (size_t)n * 128;
    ra_axis(y1, bh, (float)RA_H, b + 0,  b + 14, b + 28, b + 42);
    ra_axis(x1, bw, (float)RA_W, b + 64, b + 78, b + 92, b + 106);
}

// ---------------------------------------------------------------------------
// Kernel 3: main gather. One block per box = 64 threads = 2 waves.
// Lane owns 4 consecutive channels (v4f); wave covers 128 channels.
// All gathers are coalesced global_load_b128 (32 lanes x 16B = 512B/wave).
// ---------------------------------------------------------------------------
__device__ __forceinline__ v4f ra_ld4(const float* p) {
    return *(const v4f*)p;
}
// r = wa*a + wb*b (per component)
__device__ __forceinline__ v4f ra_lerp4(const v4f a, const v4f b,
                                        const float wa, const float wb) {
    v4f r;
    r.x = fmaf(wa, a.x, wb * b.x);
    r.y = fmaf(wa, a.y, wb * b.y);
    r.z = fmaf(wa, a.z, wb * b.z);
    r.w = fmaf(wa, a.w, wb * b.w);
    return r;
}
// acc += w*v
__device__ __forceinline__ void ra_acc4(v4f& acc, const float w, const v4f v) {
    acc.x = fmaf(w, v.x, acc.x);
    acc.y = fmaf(w, v.y, acc.y);
    acc.z = fmaf(w, v.z, acc.z);
    acc.w = fmaf(w, v.w, acc.w);
}

__global__ __launch_bounds__(64) void ra_main(const float* __restrict__ feat_hwc,
                                              const unsigned* __restrict__ blobs,
                                              float* __restrict__ out) {
    __shared__ unsigned sBlob[128];            // 512 B sampling table
    __shared__ float    sTile[2][128 * 50];    // 2 waves x (128 ch, pitch 50)

    const int n    = blockIdx.x;
    const int lane = threadIdx.x & 31;
    const int wv   = threadIdx.x >> 5;

    // ---- CDNA5 async staging: wave 0, 32 lanes x B128 = whole blob ----
    if (threadIdx.x < 32) {
        const unsigned ldsOff = (unsigned)(size_t)(&sBlob[0]) + (unsigned)(lane * 16);
        const unsigned vOff   = (unsigned)(lane * 16);
        const unsigned* gbase = blobs + (size_t)n * 128;
        asm volatile(
            "global_load_async_to_lds_b128 %0, %1, %2\n\t"
            "s_wait_asynccnt 0"
            :
            : "v"(ldsOff), "v"(vOff), "s"(gbase)
            : "memory");
    }
    __syncthreads();

    // lane -> 4 consecutive channels; wave -> 128 channels
    const int cb = (wv << 7) | (lane << 2);
    const float* Fc = feat_hwc + cb;
    float* tB = &sTile[wv][(lane << 2) * 50];  // this lane's 4 channel rows

    for (int py = 0; py < RA_P; ++py) {
        const int sy0 = 2 * py, sy1 = 2 * py + 1;
        const int   yl0 = (int)sBlob[sy0]      * (RA_W * RA_C);
        const int   yh0 = (int)sBlob[14 + sy0] * (RA_W * RA_C);
        const int   yl1 = (int)sBlob[sy1]      * (RA_W * RA_C);
        const int   yh1 = (int)sBlob[14 + sy1] * (RA_W * RA_C);
        const float wyl0 = __uint_as_float(sBlob[28 + sy0]);
        const float wyh0 = __uint_as_float(sBlob[42 + sy0]);
        const float wyl1 = __uint_as_float(sBlob[28 + sy1]);
        const float wyh1 = __uint_as_float(sBlob[42 + sy1]);
        const float* rL0 = Fc + yl0;
        const float* rH0 = Fc + yh0;
        const float* rL1 = Fc + yl1;
        const float* rH1 = Fc + yh1;

        for (int px = 0; px < RA_P; ++px) {
            v4f acc = {0.0f, 0.0f, 0.0f, 0.0f};
#pragma unroll
            for (int u = 0; u < RA_SR; ++u) {
                const int sx = 2 * px + u;
                const int xl = (int)sBlob[64 + sx] * RA_C;
                const int xh = (int)sBlob[78 + sx] * RA_C;
                const float wxl = __uint_as_float(sBlob[92 + sx]);
                const float wxh = __uint_as_float(sBlob[106 + sx]);
                // sample row pair 0 (sy0)
                ra_acc4(acc, wyl0, ra_lerp4(ra_ld4(rL0 + xl), ra_ld4(rL0 + xh), wxl, wxh));
                ra_acc4(acc, wyh0, ra_lerp4(ra_ld4(rH0 + xl), ra_ld4(rH0 + xh), wxl, wxh));
                // sample row pair 1 (sy1)
                ra_acc4(acc, wyl1, ra_lerp4(ra_ld4(rL1 + xl), ra_ld4(rL1 + xh), wxl, wxh));
                ra_acc4(acc, wyh1, ra_lerp4(ra_ld4(rH1 + xl), ra_ld4(rH1 + xh), wxl, wxh));
            }
            const int bin = py * RA_P + px;
            tB[bin]       = acc.x * 0.25f;     // mean over 2x2 samples
            tB[50 + bin]  = acc.y * 0.25f;
            tB[100 + bin] = acc.z * 0.25f;
            tB[150 + bin] = acc.w * 0.25f;
        }
    }
    __syncthreads();

    // ---- coalesced NT writeback: wave owns contiguous 128*49 floats ----
    float* obase = out + (size_t)n * (RA_C * RA_BINS) + (size_t)wv * (128 * RA_BINS);
    const float* tw = &sTile[wv][0];
    for (int j = 0; j < 49; ++j) {             // 6272 floats / (32 lanes * 4)
        const int i = (j * 32 + lane) * 4;
        v4f v;
        {
            int i0 = i + 0, c0 = i0 / RA_BINS; v.x = tw[c0 * 50 + (i0 - c0 * RA_BINS)];
            int i1 = i + 1, c1 = i1 / RA_BINS; v.y = tw[c1 * 50 + (i1 - c1 * RA_BINS)];
            int i2 = i + 2, c2 = i2 / RA_BINS; v.z = tw[c2 * 50 + (i2 - c2 * RA_BINS)];
            int i3 = i + 3, c3 = i3 / RA_BINS; v.w = tw[c3 * 50 + (i3 - c3 * RA_BINS)];
        }
        __builtin_nontemporal_store(v, (v4f*)(obase + i));
    }
}

// ---------------------------------------------------------------------------
// Host launcher.
// d_in[0]: feature_map (1,256,80,80) fp32 ; d_in[1]: proposals (N,4) fp32
// d_out:   (N,256,7,7) fp32
// d_ws:    [0, 6553600) HWC features ; [6553600, +512*N) per-box blobs
// ---------------------------------------------------------------------------
extern "C" void kernel_launch(void* const* d_in, const int* in_sizes, int n_in,
                              void* d_out, int out_size, void* d_ws, size_t ws_size,
                              hipStream_t stream) {
    const float* fmap  = (const float*)d_in[0];
    const float* props = (const float*)d_in[1];
    float* out = (float*)d_out;

    const int n_boxes = in_sizes[1] / 4;   // 1000

    float*    feat_hwc = (float*)d_ws;
    unsigned* blobs    = (unsigned*)((char*)d_ws + (size_t)RA_C * RA_HW * sizeof(float));

    ra_chw_to_hwc<<<dim3(RA_HW / 32, RA_C / 32), dim3(32, 8), 0, stream>>>(fmap, feat_hwc);
    ra_coords<<<(n_boxes + 127) / 128, 128, 0, stream>>>(props, blobs, n_boxes);
    ra_main<<<n_boxes, 64, 0, stream>>>(feat_hwc, blobs, out);
}